// MeanAggregator_91053306675295
// MI455X (gfx1250) — compile-verified
//
#include <hip/hip_runtime.h>

typedef float v2f __attribute__((ext_vector_type(2)));
typedef float v8f __attribute__((ext_vector_type(8)));

#define EMBED_DIM  512
#define NUM_SAMPLE 32
#define ROW_PAD    516   // 512 + 4 floats: keeps 16B row alignment, shifts LDS banks per row

__global__ __launch_bounds__(256)
void mean_agg_wmma_kernel(const float* __restrict__ features,
                          const float* __restrict__ neigh_weight,
                          const int*   __restrict__ neigh_idx,
                          float*       __restrict__ out)
{
    __shared__ float feat[NUM_SAMPLE][ROW_PAD];   // 66048 B gathered tile
    __shared__ float wsh[NUM_SAMPLE];

    const int b = blockIdx.x;
    const int t = threadIdx.x;          // 0..255, 8 waves of 32

    // ---- Phase 1: async gather of 32 feature rows (2KB each) into LDS ----
    {
        const int s    = t >> 3;        // sample 0..31
        const int part = t & 7;         // 8 threads cooperate per row
        const long long row = (long long)neigh_idx[b * NUM_SAMPLE + s];
        const float* gsrc = features + (size_t)row * EMBED_DIM + part * 64;
        const float* ldst = &feat[s][part * 64];
        const unsigned lds_base = (unsigned)(unsigned long long)ldst; // low 32 bits = LDS byte addr
        #pragma unroll
        for (int i = 0; i < 16; ++i) {  // 16 x b128 = 256B per thread
            unsigned lofs = lds_base + i * 16;
            const float* g = gsrc + i * 4;
            asm volatile("global_load_async_to_lds_b128 %0, %1, off"
                         :: "v"(lofs), "v"(g)
                         : "memory");
        }
    }

    // ---- Phase 2: weights to LDS, row-normalize ----
    if (t < NUM_SAMPLE) wsh[t] = neigh_weight[b * NUM_SAMPLE + t];
    __syncthreads();
    float sum = 0.f;
    #pragma unroll
    for (int s = 0; s < NUM_SAMPLE; ++s) sum += wsh[s];
    const float rsum = 1.0f / sum;

    // wait for this wave's async copies; barrier covers the other waves' copies
    asm volatile("s_wait_asynccnt 0" ::: "memory");
    __syncthreads();

    // ---- Phase 3: WMMA accumulation. One batch row, weights replicated over M.
    // A(16x4) layout: lanes 0-15 hold K=0(v0),K=1(v1); lanes 16-31 hold K=2(v0),K=3(v1).
    // B(4x16) assumed mirror: v0 = {K=0 | K=2}, v1 = {K=1 | K=3}, N = lane%16.
    const int wv   = t >> 5;            // wave 0..7
    const int lane = t & 31;
    const int m    = lane & 15;         // N column (and output column for store)
    const int hi   = lane >> 4;         // 0: K-pair {0,1}, 1: K-pair {2,3}

    #pragma unroll
    for (int i = 0; i < 4; ++i) {
        const int dt = wv * 4 + i;      // dtile 0..31 (16 dims each)
        v8f c = {0.f, 0.f, 0.f, 0.f, 0.f, 0.f, 0.f, 0.f};
        #pragma unroll
        for (int j = 0; j < 8; ++j) {   // 8 K-chunks of 4 samples
            const int k0 = j * 4 + hi * 2;
            v2f a, bm;
            a.x  = wsh[k0]     * rsum;  // weight replicated across all 16 M rows
            a.y  = wsh[k0 + 1] * rsum;
            bm.x = feat[k0    ][dt * 16 + m];
            bm.y = feat[k0 + 1][dt * 16 + m];
            c = __builtin_amdgcn_wmma_f32_16x16x4_f32(
                    false, a, false, bm, (short)0, c, false, false);
        }
        // every row of C is identical; take M=0 -> VGPR0 of lanes 0..15
        if (lane < 16) out[b * EMBED_DIM + dt * 16 + lane] = c[0];
    }
}

extern "C" void kernel_launch(void* const* d_in, const int* in_sizes, int n_in,
                              void* d_out, int out_size, void* d_ws, size_t ws_size,
                              hipStream_t stream) {
    const float* features     = (const float*)d_in[0];
    const float* neigh_weight = (const float*)d_in[1];
    const int*   neigh_idx    = (const int*)d_in[2];
    float*       out          = (float*)d_out;

    const int batch = in_sizes[1] / NUM_SAMPLE;   // neigh_weight is [B, S]
    dim3 grid(batch), block(256);
    hipLaunchKernelGGL(mean_agg_wmma_kernel, grid, block, 0, stream,
                       features, neigh_weight, neigh_idx, out);
}